// LatentMapper2_14577119002873
// MI455X (gfx1250) — compile-verified
//
#include <hip/hip_runtime.h>
#include <math.h>

// Problem constants from the reference
#define B_    32
#define T_    2048
#define TOUT_ 512
#define D_    256
#define LN_EPS 1e-5f

#define ROWS  16      // output rows (positions) per workgroup
#define PITCH 260     // padded LDS row pitch (floats) to spread banks

typedef __attribute__((ext_vector_type(2))) float v2f;
typedef __attribute__((ext_vector_type(8))) float v8f;

// Fused: ragged adaptive avg-pool -> fp32 WMMA GEMM (1x1 conv) -> exact GELU
// -> LayerNorm.  One block = 16 wave32s; block handles 16 rows x 256 cols.
__global__ __launch_bounds__(512, 1)
void latent_mapper2_fused(const float* __restrict__ mu_mel,      // [B, T, D]
                          const int*   __restrict__ mel_lengths, // [B]
                          const int*   __restrict__ target_lens, // [B]
                          const float* __restrict__ conv_w,      // [D, D] (e, d)
                          const float* __restrict__ gamma,       // [D]
                          const float* __restrict__ beta,        // [D]
                          float*       __restrict__ out)         // [B, TOUT, D]
{
    __shared__ float As[ROWS][PITCH];   // pooled rows (A matrix, M x K)
    __shared__ float Hs[ROWS][PITCH];   // GELU(h) rows for LayerNorm

    const int tid  = threadIdx.x;
    const int lane = tid & 31;
    const int wave = tid >> 5;                    // 0..15

    const int tiles_per_b = TOUT_ / ROWS;         // 32
    const int b  = blockIdx.x / tiles_per_b;
    const int j0 = (blockIdx.x % tiles_per_b) * ROWS;

    const int L = mel_lengths[b];
    const int O = target_lens[b];

    // -------- Stage 1: ragged adaptive average pool (1 wave per row) -------
    {
        const int j = j0 + wave;
        if (j < O) {
            const int start = (j * L) / O;
            const int end   = ((j + 1) * L + O - 1) / O;   // ceil
            int cnt = end - start; if (cnt < 1) cnt = 1;
            const float inv = 1.0f / (float)cnt;
            const float* src = mu_mel + ((size_t)b * T_ + (size_t)start) * D_;
            float acc[8];
            #pragma unroll
            for (int i = 0; i < 8; ++i) acc[i] = 0.0f;
            for (int t = start; t < end; ++t) {
                #pragma unroll
                for (int i = 0; i < 8; ++i)
                    acc[i] += src[i * 32 + lane];          // coalesced over D
                src += D_;
            }
            #pragma unroll
            for (int i = 0; i < 8; ++i)
                As[wave][i * 32 + lane] = acc[i] * inv;
        } else {
            #pragma unroll
            for (int i = 0; i < 8; ++i)
                As[wave][i * 32 + lane] = 0.0f;            // padded rows
        }
    }
    __syncthreads();

    // -------- Stage 2: GEMM h = A * W^T via V_WMMA_F32_16X16X4_F32 --------
    // Wave `wave` computes the 16x16 tile covering output channels
    // [wave*16, wave*16+16).  K loop over D_=256 in steps of 4.
    const int half = lane >> 4;                  // 0: lanes 0-15, 1: 16-31
    const int nl   = lane & 15;
    const int n0   = wave * 16;
    const float* wrow = conv_w + (size_t)(n0 + nl) * D_;   // W[e, :], K-contig

    v8f c = {};
    #pragma unroll 4
    for (int k = 0; k < D_; k += 4) {
        const int kk = k + half * 2;             // ISA A layout: half picks K+2,K+3
        v2f a;                                   // A 16x4: M=nl, K=kk..kk+1
        a.x = As[nl][kk];
        a.y = As[nl][kk + 1];
        v2f bb = *(const v2f*)(wrow + kk);       // B 4x16: N=nl, K=kk..kk+1 (b64 load)
        c = __builtin_amdgcn_wmma_f32_16x16x4_f32(
                /*neg_a=*/false, a, /*neg_b=*/false, bb,
                /*c_mod=*/(short)0, c, /*reuse_a=*/false, /*reuse_b=*/false);
    }

    // Exact GELU on the accumulator tile; scatter to LDS per C/D VGPR layout:
    // VGPR r holds M=r (lanes 0-15) and M=r+8 (lanes 16-31), N = nl.
    #pragma unroll
    for (int r = 0; r < 8; ++r) {
        const float x = c[r];
        const float g = 0.5f * x * (1.0f + erff(x * 0.70710678118654752f));
        Hs[r + half * 8][n0 + nl] = g;
    }
    __syncthreads();

    // -------- Stage 3: LayerNorm over D (1 wave per row) -------------------
    {
        float s1 = 0.0f, s2 = 0.0f;
        #pragma unroll
        for (int i = 0; i < 8; ++i) {
            const float h = Hs[wave][i * 32 + lane];
            s1 += h;
            s2 += h * h;
        }
        #pragma unroll
        for (int off = 16; off > 0; off >>= 1) {  // wave32 butterfly reduce
            s1 += __shfl_xor(s1, off, 32);
            s2 += __shfl_xor(s2, off, 32);
        }
        const float mean = s1 * (1.0f / (float)D_);
        float var = s2 * (1.0f / (float)D_) - mean * mean;
        if (var < 0.0f) var = 0.0f;
        const float rstd = rsqrtf(var + LN_EPS);

        float* dst = out + ((size_t)b * TOUT_ + (size_t)(j0 + wave)) * D_;
        #pragma unroll
        for (int i = 0; i < 8; ++i) {
            const int d = i * 32 + lane;
            const float h = Hs[wave][d];
            dst[d] = (h - mean) * rstd * gamma[d] + beta[d];
        }
    }
}

extern "C" void kernel_launch(void* const* d_in, const int* in_sizes, int n_in,
                              void* d_out, int out_size, void* d_ws, size_t ws_size,
                              hipStream_t stream) {
    (void)in_sizes; (void)n_in; (void)out_size; (void)d_ws; (void)ws_size;
    const float* mu_mel      = (const float*)d_in[0];
    const int*   mel_lengths = (const int*)  d_in[1];
    const int*   target_lens = (const int*)  d_in[2];
    const float* conv_w      = (const float*)d_in[3];
    const float* gamma       = (const float*)d_in[4];
    const float* beta        = (const float*)d_in[5];
    float* out = (float*)d_out;

    const int blocks = (B_ * TOUT_) / ROWS;   // 1024
    latent_mapper2_fused<<<blocks, 512, 0, stream>>>(
        mu_mel, mel_lengths, target_lens, conv_w, gamma, beta, out);
}